// RelToAbsIndex_53145925321409
// MI455X (gfx1250) — compile-verified
//
#include <hip/hip_runtime.h>
#include <hip/hip_bf16.h>

// RelToAbsIndex: elementwise int32 map over 16.78M pixels.
// HBM-bound (192 MB / call -> ~8.6 us floor at 23.3 TB/s). Strategy:
// 128-bit (native vec4) non-temporal loads/stores, ~10 cheap VALU ops/elem.

#define NW 32
#define NH 32

// Native clang vector type: accepted by __builtin_nontemporal_load/store
// (HIP's int4 is a class and is rejected).
typedef int v4i __attribute__((ext_vector_type(4)));

__device__ __forceinline__ int rel_to_abs_one(int rel, int init) {
    // rel in [0,8]: dy = rel/3 - 1, dx = rel%3 - 1.
    // rel/3 via (rel*11)>>5 (exact for 0..8), avoids 32-bit magic divide.
    int q  = (rel * 11) >> 5;
    int dx = rel - q * 3 - 1;
    int dy = q - 1;
    // NW == 32, NH == 32 -> bitwise decompose of init index.
    int gx = init & (NW - 1);
    int gy = init >> 5;
    // clamp to grid; lowers to v_med3_i32
    int x = gx + dx; x = x < 0 ? 0 : (x > NW - 1 ? NW - 1 : x);
    int y = gy + dy; y = y < 0 ? 0 : (y > NH - 1 ? NH - 1 : y);
    return (y << 5) | x;
}

__global__ __launch_bounds__(256) void RelToAbsIndex_kernel(
        const int* __restrict__ rel_map,
        const int* __restrict__ init_map,
        int* __restrict__ out,
        int n /* total elements */) {
    const int n4 = n >> 2;                       // number of full 4-wide packets
    const int i  = blockIdx.x * blockDim.x + threadIdx.x;

    if (i < n4) {
        // 128-bit non-temporal streaming path (global_load_b128, TH=NT):
        // data is touched exactly once, don't retain in L2.
        const v4i* __restrict__ r4 = (const v4i*)rel_map;
        const v4i* __restrict__ g4 = (const v4i*)init_map;
        v4i*       __restrict__ o4 = (v4i*)out;

        v4i r = __builtin_nontemporal_load(&r4[i]);
        v4i g = __builtin_nontemporal_load(&g4[i]);
        v4i o;
        o.x = rel_to_abs_one(r.x, g.x);
        o.y = rel_to_abs_one(r.y, g.y);
        o.z = rel_to_abs_one(r.z, g.z);
        o.w = rel_to_abs_one(r.w, g.w);
        __builtin_nontemporal_store(o, &o4[i]);
    }

    // Scalar tail (n % 4 != 0). Not taken for this problem size (16,777,216),
    // but keeps the kernel correct for any n. Handled by thread i == n4.
    if (i == n4) {
        for (int k = n4 << 2; k < n; ++k) {
            int r = rel_map[k];
            int g = init_map[k];
            out[k] = rel_to_abs_one(r, g);
        }
    }
}

extern "C" void kernel_launch(void* const* d_in, const int* in_sizes, int n_in,
                              void* d_out, int out_size, void* d_ws, size_t ws_size,
                              hipStream_t stream) {
    (void)n_in; (void)d_ws; (void)ws_size; (void)out_size;

    const int* rel_map  = (const int*)d_in[0];   // rel_idx_map, int32
    const int* init_map = (const int*)d_in[1];   // init_idx_map, int32
    int*       out      = (int*)d_out;           // int32 output (4B elems)

    const int n  = in_sizes[0];                  // 64*512*512 = 16,777,216
    const int n4 = n >> 2;                       // one vec4 per thread
    // +1 thread slot so the tail handler (thread id == n4) always exists.
    const int threads = 256;
    const int blocks  = (n4 + 1 + threads - 1) / threads;

    RelToAbsIndex_kernel<<<blocks, threads, 0, stream>>>(rel_map, init_map, out, n);
}